// impactDetect_3393024164038
// MI455X (gfx1250) — compile-verified
//
#include <hip/hip_runtime.h>

#define NSL_ATT 0.2f
#define NSL_MLP 0.01f

typedef __attribute__((ext_vector_type(2))) float v2f;
typedef __attribute__((ext_vector_type(8))) float v8f;

#define CDIV(a, b) (((a) + (b) - 1) / (b))

// ---------- helpers ----------

// order-preserving float -> uint key (for atomic float max via atomicMax(u32))
__device__ __forceinline__ unsigned fkey(float f) {
  unsigned u = __float_as_uint(f);
  return (u & 0x80000000u) ? ~u : (u | 0x80000000u);
}
__device__ __forceinline__ float funkey(unsigned k) {
  return __uint_as_float((k & 0x80000000u) ? (k ^ 0x80000000u) : ~k);
}

__device__ __forceinline__ void edge_sd(int i, const int* __restrict__ src,
                                        const int* __restrict__ dst, int nE,
                                        int& s, int& d) {
  if (i < nE) { s = src[i]; d = dst[i]; }
  else        { s = i - nE; d = s; }        // implicit self-loop
}

// ---------- kernels ----------

__global__ void k_fill_u32(unsigned* __restrict__ p, unsigned v, int n) {
  int i = blockIdx.x * blockDim.x + threadIdx.x;
  if (i < n) p[i] = v;
}

// H[nrows x 64] = A[nrows x K] @ B[K x 64]   (fp32, V_WMMA_F32_16X16X4_F32)
// One wave32 computes one 16x16 output tile. nrows % 16 == 0, K % 4 == 0.
__global__ void k_gemm_wmma_f32(const float* __restrict__ A,
                                const float* __restrict__ B,
                                float* __restrict__ H, int nrows, int K) {
  int gt   = blockIdx.x * blockDim.x + threadIdx.x;
  int wave = gt >> 5;
  int lane = threadIdx.x & 31;
  int colTile = wave & 3;        // 64 cols = 4 tiles of 16
  int rowTile = wave >> 2;
  if (rowTile * 16 >= nrows) return;   // wave-uniform exit (EXEC all-ones for WMMA)

  int r0 = rowTile * 16;
  int c0 = colTile * 16;
  int mrow  = lane & 15;           // row within tile (A) / col within tile (B,D)
  int khalf = (lane >> 4) * 2;     // lanes 16..31 hold K+2,K+3 halves

  v8f acc = {};
  for (int k = 0; k < K; k += 4) {
    // A 16x4 fp32 fragment: lane holds A[r0+mrow][k+khalf], A[r0+mrow][k+khalf+1]
    const float* ap = A + (size_t)(r0 + mrow) * K + (k + khalf);
    v2f a; a.x = ap[0]; a.y = ap[1];
    // B 4x16 fp32 fragment: VGPR0 = row (k+khalf), VGPR1 = row (k+khalf+1), N = mrow
    v2f b;
    b.x = B[(size_t)(k + khalf)     * 64 + c0 + mrow];
    b.y = B[(size_t)(k + khalf + 1) * 64 + c0 + mrow];
    acc = __builtin_amdgcn_wmma_f32_16x16x4_f32(
        /*neg_a=*/false, a, /*neg_b=*/false, b,
        /*c_mod=*/(short)0, acc, /*reuse_a=*/false, /*reuse_b=*/false);
  }

  // D layout: VGPR i -> M = i (+8 for upper lane half), N = lane&15
  int ncol  = c0 + mrow;
  int mbase = r0 + ((lane >> 4) << 3);
#pragma unroll
  for (int i = 0; i < 8; ++i)
    H[(size_t)(mbase + i) * 64 + ncol] = acc[i];
}

// per-node attention logits: as[i] = h[i,:].a_src ; ad[i] = h[i,:].a_dst
__global__ void k_alphas(const float* __restrict__ h, const float* __restrict__ a_src,
                         const float* __restrict__ a_dst, float* __restrict__ as,
                         float* __restrict__ ad, int n) {
  int i = blockIdx.x * blockDim.x + threadIdx.x;
  if (i >= n) return;
  const float* hp = h + (size_t)i * 64;
  float s = 0.f, d = 0.f;
#pragma unroll 8
  for (int c = 0; c < 64; ++c) { float v = hp[c]; s += v * a_src[c]; d += v * a_dst[c]; }
  as[i] = s; ad[i] = d;
}

// pass A: per-dst segment max of leaky_relu(as[src]+ad[dst])
__global__ void k_edge_max(const float* __restrict__ as, const float* __restrict__ ad,
                           const int* __restrict__ src, const int* __restrict__ dst,
                           unsigned* __restrict__ mkey, int nE, int nN) {
  int i = blockIdx.x * blockDim.x + threadIdx.x;
  if (i >= nE + nN) return;
  int s, d; edge_sd(i, src, dst, nE, s, d);
  float e = as[s] + ad[d];
  e = (e > 0.f) ? e : NSL_ATT * e;
  atomicMax(&mkey[d], fkey(e));
}

// pass B: e2 = exp(e - max[dst]); denom[dst] += e2; stash e2 per edge
__global__ void k_edge_denom(const float* __restrict__ as, const float* __restrict__ ad,
                             const int* __restrict__ src, const int* __restrict__ dst,
                             const unsigned* __restrict__ mkey, float* __restrict__ denom,
                             float* __restrict__ ealpha, int nE, int nN) {
  int i = blockIdx.x * blockDim.x + threadIdx.x;
  if (i >= nE + nN) return;
  int s, d; edge_sd(i, src, dst, nE, s, d);
  float e = as[s] + ad[d];
  e = (e > 0.f) ? e : NSL_ATT * e;
  float e2 = __expf(e - funkey(mkey[d]));
  ealpha[i] = e2;
  atomicAdd(&denom[d], e2);
}

// pass C: acc[dst,c] += h[src,c] * e2/(denom[dst]+1e-16); one thread per (edge, channel)
__global__ void k_edge_scatter(const float* __restrict__ h, const float* __restrict__ ealpha,
                               const float* __restrict__ denom, const int* __restrict__ src,
                               const int* __restrict__ dst, float* __restrict__ acc,
                               int nE, int nN) {
  unsigned t = blockIdx.x * blockDim.x + threadIdx.x;
  int i = (int)(t >> 6);
  int c = (int)(t & 63u);
  if (i >= nE + nN) return;
  int s, d; edge_sd(i, src, dst, nE, s, d);
  float alpha = ealpha[i] / (denom[d] + 1e-16f);
  atomicAdd(&acc[(size_t)d * 64 + c], h[(size_t)s * 64 + c] * alpha);
}

// Z[i,c] += b[c]; optional ReLU (layer 1)
__global__ void k_bias_act(float* __restrict__ Z, const float* __restrict__ b,
                           int n, int relu) {
  int t = blockIdx.x * blockDim.x + threadIdx.x;
  if (t >= n * 64) return;
  float v = Z[t] + b[t & 63];
  Z[t] = relu ? fmaxf(v, 0.f) : v;
}

// out[i] = leaky_relu( Z[idx[i],:].w + b0 , 0.01 )
__global__ void k_head_scalar(const float* __restrict__ Z, const int* __restrict__ idx,
                              const float* __restrict__ w, const float* __restrict__ b,
                              float* __restrict__ out, int n) {
  int i = blockIdx.x * blockDim.x + threadIdx.x;
  if (i >= n) return;
  const float* z = Z + (size_t)idx[i] * 64;
  float a = b[0];
#pragma unroll 8
  for (int c = 0; c < 64; ++c) a += z[c] * w[c];
  out[i] = (a > 0.f) ? a : NSL_MLP * a;
}

// out[i,0:2] = Z[i,:] @ Wb + bb   (no activation: GRL is identity in fwd)
__global__ void k_fprob(const float* __restrict__ Z, const float* __restrict__ Wb,
                        const float* __restrict__ bb, float* __restrict__ out, int n) {
  int i = blockIdx.x * blockDim.x + threadIdx.x;
  if (i >= n) return;
  const float* z = Z + (size_t)i * 64;
  float a0 = bb[0], a1 = bb[1];
#pragma unroll 8
  for (int c = 0; c < 64; ++c) { float v = z[c]; a0 += v * Wb[c * 2]; a1 += v * Wb[c * 2 + 1]; }
  out[i * 2] = a0; out[i * 2 + 1] = a1;
}

// tprob tail: T1 holds Z@Wp1 (from WMMA GEMM); apply bp1 + lrelu, then @Wp2 + bp2, lrelu
__global__ void k_tprob2(const float* __restrict__ T1, const float* __restrict__ bp1,
                         const float* __restrict__ Wp2, const float* __restrict__ bp2,
                         float* __restrict__ out, int n) {
  int i = blockIdx.x * blockDim.x + threadIdx.x;
  if (i >= n) return;
  const float* t = T1 + (size_t)i * 64;
  float a0 = bp2[0], a1 = bp2[1];
#pragma unroll 8
  for (int c = 0; c < 64; ++c) {
    float v = t[c] + bp1[c];
    v = (v > 0.f) ? v : NSL_MLP * v;
    a0 += v * Wp2[c * 2]; a1 += v * Wp2[c * 2 + 1];
  }
  out[i * 2]     = (a0 > 0.f) ? a0 : NSL_MLP * a0;
  out[i * 2 + 1] = (a1 > 0.f) ? a1 : NSL_MLP * a1;
}

// ---------- host launch ----------

extern "C" void kernel_launch(void* const* d_in, const int* in_sizes, int n_in,
                              void* d_out, int out_size, void* d_ws, size_t ws_size,
                              hipStream_t stream) {
  const int IN = 256, HC = 64;
  const float* x      = (const float*)d_in[0];
  const float* fake_x = (const float*)d_in[1];
  const int* ei       = (const int*)d_in[2];
  const int* fei      = (const int*)d_in[3];
  const int* treat    = (const int*)d_in[4];
  const int* control  = (const int*)d_in[5];
  const float* W1     = (const float*)d_in[6];
  const float* as1    = (const float*)d_in[7];
  const float* ad1    = (const float*)d_in[8];
  const float* b1     = (const float*)d_in[9];
  const float* W2     = (const float*)d_in[10];
  const float* as2    = (const float*)d_in[11];
  const float* ad2    = (const float*)d_in[12];
  const float* b2     = (const float*)d_in[13];
  const float* Wy1    = (const float*)d_in[14];
  const float* by1    = (const float*)d_in[15];
  const float* Wy0    = (const float*)d_in[16];
  const float* by0    = (const float*)d_in[17];
  const float* Wb     = (const float*)d_in[18];
  const float* bb     = (const float*)d_in[19];
  const float* Wp1    = (const float*)d_in[20];
  const float* bp1    = (const float*)d_in[21];
  const float* Wp2    = (const float*)d_in[22];
  const float* bp2    = (const float*)d_in[23];

  const int N  = in_sizes[0] / IN;      // 50000
  const int E  = in_sizes[2] / 2;       // 800000
  const int NT = in_sizes[4];           // 5000
  const int NC = in_sizes[5];           // 5000
  const int nEdges = E + N;             // with self-loops

  // workspace layout (floats)
  float* ws = (float*)d_ws;
  float* h      = ws;                 // N*64  (GEMM output / tprob hidden)
  float* z1     = h    + (size_t)N * 64;   // N*64  (layer-1 accum -> xZ1)
  float* z2r    = z1   + (size_t)N * 64;   // N*64  (xZ2 real)
  float* z2f    = z2r  + (size_t)N * 64;   // N*64  (xZ2 fake)
  float* asbuf  = z2f  + (size_t)N * 64;   // N
  float* adbuf  = asbuf + N;               // N
  float* denom  = adbuf + N;               // N
  unsigned* mkey = (unsigned*)(denom + N); // N
  float* ealpha = (float*)(mkey + N);      // E+N

  float* out = (float*)d_out;
  float* o_y1     = out;               // 5000
  float* o_yc0    = o_y1  + NT;        // 5000
  float* o_y0     = o_yc0 + NT;        // 5000
  float* o_yc1    = o_y0  + NC;        // 5000
  float* o_fprob  = o_yc1 + NC;        // N*2
  float* o_fprobf = o_fprob + (size_t)N * 2;
  float* o_tprob  = o_fprobf + (size_t)N * 2;

  const int BT = 256;
  const int gemm_waves   = (N / 16) * 4;              // one wave per 16x16 tile
  const int gemm_blocks  = CDIV(gemm_waves * 32, BT);
  const int node_blocks  = CDIV(N, BT);
  const int feat_blocks  = CDIV(N * 64, BT);
  const int edge_blocks  = CDIV(nEdges, BT);
  const unsigned scat_threads = (unsigned)nEdges * 64u;
  const int scat_blocks  = (int)CDIV(scat_threads, (unsigned)BT);

  for (int g = 0; g < 2; ++g) {
    const float* X   = g ? fake_x : x;
    const int*   SRC = g ? fei : ei;
    const int*   DST = SRC + E;
    float* z2 = g ? z2f : z2r;

    // ---- layer 1 ----
    k_fill_u32<<<feat_blocks, BT, 0, stream>>>((unsigned*)z1, 0u, N * 64);
    k_fill_u32<<<node_blocks, BT, 0, stream>>>(mkey, 0u, N);
    k_fill_u32<<<node_blocks, BT, 0, stream>>>((unsigned*)denom, 0u, N);
    k_gemm_wmma_f32<<<gemm_blocks, BT, 0, stream>>>(X, W1, h, N, IN);
    k_alphas<<<node_blocks, BT, 0, stream>>>(h, as1, ad1, asbuf, adbuf, N);
    k_edge_max<<<edge_blocks, BT, 0, stream>>>(asbuf, adbuf, SRC, DST, mkey, E, N);
    k_edge_denom<<<edge_blocks, BT, 0, stream>>>(asbuf, adbuf, SRC, DST, mkey, denom, ealpha, E, N);
    k_edge_scatter<<<scat_blocks, BT, 0, stream>>>(h, ealpha, denom, SRC, DST, z1, E, N);
    k_bias_act<<<feat_blocks, BT, 0, stream>>>(z1, b1, N, /*relu=*/1);

    // ---- layer 2 ----
    k_fill_u32<<<feat_blocks, BT, 0, stream>>>((unsigned*)z2, 0u, N * 64);
    k_fill_u32<<<node_blocks, BT, 0, stream>>>(mkey, 0u, N);
    k_fill_u32<<<node_blocks, BT, 0, stream>>>((unsigned*)denom, 0u, N);
    k_gemm_wmma_f32<<<gemm_blocks, BT, 0, stream>>>(z1, W2, h, N, HC);
    k_alphas<<<node_blocks, BT, 0, stream>>>(h, as2, ad2, asbuf, adbuf, N);
    k_edge_max<<<edge_blocks, BT, 0, stream>>>(asbuf, adbuf, SRC, DST, mkey, E, N);
    k_edge_denom<<<edge_blocks, BT, 0, stream>>>(asbuf, adbuf, SRC, DST, mkey, denom, ealpha, E, N);
    k_edge_scatter<<<scat_blocks, BT, 0, stream>>>(h, ealpha, denom, SRC, DST, z2, E, N);
    k_bias_act<<<feat_blocks, BT, 0, stream>>>(z2, b2, N, /*relu=*/0);
  }

  // ---- heads ----
  const int ib = CDIV(NT, BT);
  k_head_scalar<<<ib, BT, 0, stream>>>(z2r, treat,   Wy1, by1, o_y1,  NT);
  k_head_scalar<<<ib, BT, 0, stream>>>(z2f, treat,   Wy0, by0, o_yc0, NT);
  k_head_scalar<<<ib, BT, 0, stream>>>(z2r, control, Wy0, by0, o_y0,  NC);
  k_head_scalar<<<ib, BT, 0, stream>>>(z2f, control, Wy1, by1, o_yc1, NC);
  k_fprob<<<node_blocks, BT, 0, stream>>>(z2r, Wb, bb, o_fprob,  N);
  k_fprob<<<node_blocks, BT, 0, stream>>>(z2f, Wb, bb, o_fprobf, N);
  // tprob: hidden = z2r @ Wp1 via WMMA, then fused bias/lrelu + 2-col tail
  k_gemm_wmma_f32<<<gemm_blocks, BT, 0, stream>>>(z2r, Wp1, h, N, HC);
  k_tprob2<<<node_blocks, BT, 0, stream>>>(h, bp1, Wp2, bp2, o_tprob, N);
}